// GELU277_23648089932120
// MI455X (gfx1250) — compile-verified
//
#include <hip/hip_runtime.h>
#include <math.h>

// ---------------------------------------------------------------------------
// GELU277 steady-state path for MI455X (gfx1250, wave32).
//
// Memory-bound: >= 268 MB HBM traffic -> ~11.5us @ 23.3 TB/s. Strategy:
//  K1: stream x (HBM->L2, 192MB L2 holds all 134MB of x), gelu + column sums.
//  K2: 1 block; mean, norm, sims = buf_keys @ m via v_wmma_f32_16x16x4_f32,
//      argmax, scalar gate math, emit gate[2048].
//  K3: re-read x (L2 hits), recompute gelu, multiply by gate (LDS-staged),
//      non-temporal stores for the 134MB output stream (don't evict x).
// ---------------------------------------------------------------------------

typedef float v2f __attribute__((ext_vector_type(2)));
typedef float v4f __attribute__((ext_vector_type(4)));
typedef float v8f __attribute__((ext_vector_type(8)));

#define DIM   2048          // D
#define NROWS 16384         // B*T
#define NBUF  512
#define RPB   32            // rows per reduce block
#define RGRP  (NROWS / RPB) // 512 row groups
#define K1_GRID (2 * RGRP)  // 2 column halves * 512 row groups = 1024 blocks
#define K3_GRID 2048

__device__ __forceinline__ float gelu_f(float v) {
  const float c0 = 0.7978845608028654f;  // sqrt(2/pi)
  float u = c0 * (v + 0.044715f * v * v * v);
  return 0.5f * v * (1.0f + tanhf(u));
}

// --- K0: zero the column-sum accumulator (must happen every call) ----------
__global__ __launch_bounds__(256) void k_zero(float* __restrict__ colsum) {
  for (int d = threadIdx.x; d < DIM; d += 256) colsum[d] = 0.0f;
}

// --- K1: gelu(x) column partial sums ---------------------------------------
// Block = 256 threads, each thread owns 4 consecutive columns (float4) and
// walks RPB rows; one atomicAdd per column at the end (512 adds/column total).
__global__ __launch_bounds__(256) void k_reduce(const float* __restrict__ x,
                                                float* __restrict__ colsum) {
  const v4f* xv = (const v4f*)x;
  int cg   = blockIdx.x & 1;        // column half
  int rg   = blockIdx.x >> 1;       // row group
  int col4 = cg * 256 + threadIdx.x;            // [0, 512) float4 column
  size_t base = (size_t)rg * RPB * (DIM / 4) + col4;
  v4f acc = {0.f, 0.f, 0.f, 0.f};
  #pragma unroll 4
  for (int r = 0; r < RPB; ++r) {
    v4f v = xv[base + (size_t)r * (DIM / 4)];
    acc.x += gelu_f(v.x);
    acc.y += gelu_f(v.y);
    acc.z += gelu_f(v.z);
    acc.w += gelu_f(v.w);
  }
  int c = col4 * 4;
  atomicAdd(&colsum[c + 0], acc.x);
  atomicAdd(&colsum[c + 1], acc.y);
  atomicAdd(&colsum[c + 2], acc.z);
  atomicAdd(&colsum[c + 3], acc.w);
}

// --- K2: mean/norm -> sims (WMMA) -> argmax -> gate vector -----------------
// One block of 512 threads (16 waves). Each wave computes 2 tiles of 16 sims
// rows with V_WMMA_F32_16X16X4_F32: A = buf_keys[16 x K4 chunk], B = m
// broadcast into all 16 N columns, C accumulates over K=2048 (4 independent
// accumulators to break the RAW chain). All 32 lanes active -> EXEC == ~0.
__global__ __launch_bounds__(512) void k_sims(const float* __restrict__ colsum,
                                              const float* __restrict__ buf_keys,
                                              const float* __restrict__ buf_masks,
                                              const float* __restrict__ facil,
                                              const unsigned char* __restrict__ valid,
                                              const float* __restrict__ log_strength,
                                              float* __restrict__ gate) {
  __shared__ float m_sh[DIM];
  __shared__ float sims_sh[NBUF];
  __shared__ float red_s[512];
  __shared__ int   red_i[512];
  __shared__ float s_invnorm;
  __shared__ float s_km1;
  __shared__ int   s_nearest;

  int tid = threadIdx.x;

  // mean vector + squared-norm reduction
  float ss = 0.f;
  for (int d = tid; d < DIM; d += 512) {
    float mv = colsum[d] * (1.0f / (float)NROWS);
    m_sh[d] = mv;
    ss += mv * mv;
  }
  red_s[tid] = ss;
  __syncthreads();
  for (int off = 256; off > 0; off >>= 1) {
    if (tid < off) red_s[tid] += red_s[tid + off];
    __syncthreads();
  }
  if (tid == 0) s_invnorm = 1.0f / sqrtf(red_s[0]);
  __syncthreads();

  // sims = buf_keys @ m via f32 WMMA (normalization by 1/||m|| applied after)
  int wave = tid >> 5;
  int lane = tid & 31;
  int am   = lane & 15;          // A: lanes 0-15 -> M, lanes 16-31 -> same M, K+2
  int kb   = (lane >> 4) << 1;   // 0 or 2
  for (int t = 0; t < 2; ++t) {
    int row0 = (wave * 2 + t) * 16;
    const float* arow = buf_keys + (size_t)(row0 + am) * DIM + kb;
    const float* brow = m_sh + kb;   // B[k][n] = m[k] for all n (broadcast)
    v8f c0 = {}, c1 = {}, c2 = {}, c3 = {};
    for (int k = 0; k < DIM; k += 16) {
      v2f a, b;
      a.x = arow[k];      a.y = arow[k + 1];
      b.x = brow[k];      b.y = brow[k + 1];
      c0 = __builtin_amdgcn_wmma_f32_16x16x4_f32(false, a, false, b, (short)0, c0, false, false);
      a.x = arow[k + 4];  a.y = arow[k + 5];
      b.x = brow[k + 4];  b.y = brow[k + 5];
      c1 = __builtin_amdgcn_wmma_f32_16x16x4_f32(false, a, false, b, (short)0, c1, false, false);
      a.x = arow[k + 8];  a.y = arow[k + 9];
      b.x = brow[k + 8];  b.y = brow[k + 9];
      c2 = __builtin_amdgcn_wmma_f32_16x16x4_f32(false, a, false, b, (short)0, c2, false, false);
      a.x = arow[k + 12]; a.y = arow[k + 13];
      b.x = brow[k + 12]; b.y = brow[k + 13];
      c3 = __builtin_amdgcn_wmma_f32_16x16x4_f32(false, a, false, b, (short)0, c3, false, false);
    }
    v8f c;
    for (int j = 0; j < 8; ++j) c[j] = (c0[j] + c1[j]) + (c2[j] + c3[j]);
    // C layout: VGPR j -> M=j (lanes 0-15, N=lane) / M=8+j (lanes 16-31).
    // Every N column holds the same dot product; read N=0 columns.
    if (lane == 0)  { for (int j = 0; j < 8; ++j) sims_sh[row0 + j]     = c[j]; }
    if (lane == 16) { for (int j = 0; j < 8; ++j) sims_sh[row0 + 8 + j] = c[j]; }
  }
  __syncthreads();

  // masked argmax (first-max tie break, like jnp.argmax)
  {
    float s = sims_sh[tid] * s_invnorm;
    if (!valid[tid]) s = -1.0f;
    red_s[tid] = s;
    red_i[tid] = tid;
  }
  __syncthreads();
  for (int off = 256; off > 0; off >>= 1) {
    if (tid < off) {
      float sa = red_s[tid],       sb = red_s[tid + off];
      int   ia = red_i[tid],       ib = red_i[tid + off];
      if (sb > sa || (sb == sa && ib < ia)) { red_s[tid] = sb; red_i[tid] = ib; }
    }
    __syncthreads();
  }
  if (tid == 0) {
    int   nearest = red_i[0];
    float smax    = red_s[0];
    float f = facil[nearest] * (smax > 0.85f ? 2.0f : 1.0f);   // FACIL_RATE
    float strength = fminf(fmaxf(expf(log_strength[0]), 0.01f), 5.0f);
    float kamp = fminf(1.0f + strength * (f - 1.0f), 8.0f);    // MAX_GATE
    s_km1 = kamp - 1.0f;
    s_nearest = nearest;
  }
  __syncthreads();
  float km1 = s_km1;
  const float* mask = buf_masks + (size_t)s_nearest * DIM;
  for (int d = tid; d < DIM; d += 512) gate[d] = 1.0f + km1 * mask[d];
}

// --- K3: out = gelu(x) * gate[col], x from L2, out via non-temporal stores -
__global__ __launch_bounds__(256) void k_apply(const float* __restrict__ x,
                                               const float* __restrict__ gate,
                                               float* __restrict__ out) {
  __shared__ float g_sh[DIM];
  for (int d = threadIdx.x; d < DIM; d += 256) g_sh[d] = gate[d];
  __syncthreads();
  const v4f* xv = (const v4f*)x;
  v4f* ov = (v4f*)out;
  size_t n4 = (size_t)NROWS * (DIM / 4);
  size_t stride = (size_t)gridDim.x * 256;
  for (size_t i = (size_t)blockIdx.x * 256 + threadIdx.x; i < n4; i += stride) {
    v4f v = xv[i];
    int c = (int)(i & (size_t)(DIM / 4 - 1)) << 2;
    v4f o;
    o.x = gelu_f(v.x) * g_sh[c + 0];
    o.y = gelu_f(v.y) * g_sh[c + 1];
    o.z = gelu_f(v.z) * g_sh[c + 2];
    o.w = gelu_f(v.w) * g_sh[c + 3];
    __builtin_nontemporal_store(o, &ov[i]);   // don't let out stream evict x in L2
  }
}

extern "C" void kernel_launch(void* const* d_in, const int* in_sizes, int n_in,
                              void* d_out, int out_size, void* d_ws, size_t ws_size,
                              hipStream_t stream) {
  const float*         x            = (const float*)d_in[0];          // (4,4096,2048)
  const float*         log_strength = (const float*)d_in[1];          // scalar
  const float*         buf_keys     = (const float*)d_in[2];          // (512,2048)
  const float*         buf_masks    = (const float*)d_in[3];          // (512,2048)
  const float*         facil        = (const float*)d_in[4];          // (512,)
  const unsigned char* valid        = (const unsigned char*)d_in[5];  // (512,) bool
  float* out    = (float*)d_out;
  float* ws     = (float*)d_ws;
  float* colsum = ws;          // 2048 f32
  float* gate   = ws + DIM;    // 2048 f32

  k_zero  <<<1,       256, 0, stream>>>(colsum);
  k_reduce<<<K1_GRID, 256, 0, stream>>>(x, colsum);
  k_sims  <<<1,       512, 0, stream>>>(colsum, buf_keys, buf_masks, facil, valid,
                                        log_strength, gate);
  k_apply <<<K3_GRID, 256, 0, stream>>>(x, gate, out);
}